// FusionAwareInterp_67980742361323
// MI455X (gfx1250) — compile-verified
//
#include <hip/hip_runtime.h>
#include <hip/hip_bf16.h>

// Problem constants (match reference)
constexpr int BS  = 2;
constexpr int NPT = 4096;   // N points
constexpr int H   = 64;
constexpr int W   = 192;
constexpr int C3D = 64;
constexpr int KNN = 3;
constexpr int HW  = H * W;  // 12288

typedef __attribute__((ext_vector_type(2))) float v2f;
typedef __attribute__((ext_vector_type(8))) float v8f;

__device__ __forceinline__ float leaky(float x) { return x >= 0.f ? x : 0.1f * x; }

// ---------------------------------------------------------------------------
// Kernel 1: brute-force top-3 nearest uv point per pixel query.
// uv for one batch (2*4096 f32 = 32KB) is staged in LDS; each thread scans all
// 4096 points keeping a sorted best-3 (strict < keeps the earliest index on
// ties, matching stable top_k).
// ---------------------------------------------------------------------------
__global__ void __launch_bounds__(256) knn_kernel(const float* __restrict__ uv,
                                                  int* __restrict__ knn_idx) {
    __shared__ float sux[NPT];
    __shared__ float suy[NPT];
    const int g = blockIdx.x * 256 + threadIdx.x;   // global query id
    const int b = g / HW;                           // HW % 256 == 0: no straddle
    const int q = g % HW;
    const float* uvb = uv + (size_t)b * 2 * NPT;
    for (int i = threadIdx.x; i < NPT; i += 256) {
        sux[i] = uvb[i];
        suy[i] = uvb[NPT + i];
    }
    __syncthreads();

    const float gx = (float)(q % W);
    const float gy = (float)(q / W);

    float d0 = 3.4e38f, d1 = 3.4e38f, d2 = 3.4e38f;
    int   i0 = 0,       i1 = 0,       i2 = 0;
    for (int n = 0; n < NPT; ++n) {
        const float dx = sux[n] - gx;
        const float dy = suy[n] - gy;
        const float d  = __fmaf_rn(dx, dx, dy * dy);
        if (d < d2) {
            if (d < d0)       { d2 = d1; i2 = i1; d1 = d0; i1 = i0; d0 = d; i0 = n; }
            else if (d < d1)  { d2 = d1; i2 = i1; d1 = d;  i1 = n; }
            else              { d2 = d;  i2 = n; }
        }
    }
    knn_idx[(size_t)g * KNN + 0] = i0;
    knn_idx[(size_t)g * KNN + 1] = i1;
    knn_idx[(size_t)g * KNN + 2] = i2;
}

// ---------------------------------------------------------------------------
// Kernel 2: gather + score MLP + weighted neighbor sum -> final (bs, C3D, HW).
// One thread per query. h1 for all 3 neighbors computed first (48 regs), then
// per output channel the sigmoid score and the weighted feat_3d accumulation.
// All weight reads are wave-uniform (scalar-cache friendly); feat_3d gathers
// hit L2 (feat_3d is 2MB).
// ---------------------------------------------------------------------------
__global__ void __launch_bounds__(256) mlp_gather_kernel(
    const float* __restrict__ uv, const float* __restrict__ feat3d,
    const float* __restrict__ w1, const float* __restrict__ b1,
    const float* __restrict__ w2, const float* __restrict__ b2,
    const int* __restrict__ knn_idx, float* __restrict__ fin) {
    const int g = blockIdx.x * 256 + threadIdx.x;
    const int b = g / HW;
    const int q = g % HW;
    const float gx = (float)(q % W);
    const float gy = (float)(q / W);
    const float* uvb = uv + (size_t)b * 2 * NPT;

    int   idx[KNN];
    float h1[KNN][16];
    for (int k = 0; k < KNN; ++k) {
        const int n = knn_idx[(size_t)g * KNN + k];
        idx[k] = n;
        const float dx  = uvb[n] - gx;
        const float dy  = uvb[NPT + n] - gy;
        const float nrm = sqrtf(__fmaf_rn(dx, dx, dy * dy));
        #pragma unroll
        for (int j = 0; j < 16; ++j) {
            float a = b1[j];
            a = __fmaf_rn(w1[j * 3 + 0], dx,  a);
            a = __fmaf_rn(w1[j * 3 + 1], dy,  a);
            a = __fmaf_rn(w1[j * 3 + 2], nrm, a);
            h1[k][j] = leaky(a);
        }
    }

    const float* f3b = feat3d + (size_t)b * C3D * NPT;
    float* finb = fin + (size_t)b * C3D * HW;
    for (int c = 0; c < C3D; ++c) {
        const float* w2c = w2 + c * 16;
        float acc = 0.f;
        #pragma unroll
        for (int k = 0; k < KNN; ++k) {
            float s = b2[c];
            #pragma unroll
            for (int j = 0; j < 16; ++j) s = __fmaf_rn(w2c[j], h1[k][j], s);
            s = 1.f / (1.f + __expf(-s));                 // sigmoid
            acc = __fmaf_rn(s, f3b[(size_t)c * NPT + idx[k]], acc);
        }
        finb[(size_t)c * HW + q] = acc;                   // coalesced over q
    }
}

// ---------------------------------------------------------------------------
// Kernel 3: out = leaky(w_out @ final + b_out) via V_WMMA_F32_16X16X4_F32.
// Per wave: one 16x16 output tile (M = out channels, N = pixels), K=64 in 16
// WMMA steps. A-frag (16x4 f32, 2 VGPRs): lane l -> M=l&15, K=(l>>4)*2+{0,1}.
// B-frag (4x16 f32, 2 VGPRs): lane l -> N=l&15, K=(l>>4)*2+{0,1}.
// C/D (16x16 f32, 8 VGPRs): lane l -> N=l&15, M=r+((l>=16)?8:0).
// Grid sized so EXEC is all-1s in every wave (WMMA requirement).
// ---------------------------------------------------------------------------
__global__ void __launch_bounds__(256) proj_wmma_kernel(
    const float* __restrict__ fin, const float* __restrict__ w_out,
    const float* __restrict__ b_out, float* __restrict__ out) {
    constexpr int QT = HW / 16;                 // 768 q-tiles
    constexpr int OT = C3D / 16;                // 4   o-tiles
    const int wave = (blockIdx.x * 256 + threadIdx.x) >> 5;
    const int lane = threadIdx.x & 31;
    const int qt = wave % QT;
    const int ot = (wave / QT) % OT;
    const int b  = wave / (QT * OT);

    const int mn = lane & 15;                   // M for A-frag, N for B/D-frag
    const int kb = (lane >> 4) * 2;             // K sub-offset for lane half

    const float* Arow = w_out + (size_t)(ot * 16 + mn) * C3D;           // w_out[o][*]
    const float* Bcol = fin + (size_t)b * C3D * HW + (size_t)qt * 16 + mn;

    v8f acc = {};
    #pragma unroll
    for (int k0 = 0; k0 < C3D; k0 += 4) {
        v2f a, bb;
        a.x  = Arow[k0 + kb];
        a.y  = Arow[k0 + kb + 1];
        bb.x = Bcol[(size_t)(k0 + kb) * HW];
        bb.y = Bcol[(size_t)(k0 + kb + 1) * HW];
        acc = __builtin_amdgcn_wmma_f32_16x16x4_f32(
            /*neg_a=*/false, a, /*neg_b=*/false, bb,
            /*c_mod=*/(short)0, acc, /*reuse_a=*/false, /*reuse_b=*/false);
    }

    float* outb = out + (size_t)b * C3D * HW;
    const int mhalf = (lane >> 4) * 8;
    #pragma unroll
    for (int r = 0; r < 8; ++r) {
        const int o = ot * 16 + r + mhalf;
        float v = acc[r] + b_out[o];
        v = leaky(v);
        outb[(size_t)o * HW + (size_t)qt * 16 + mn] = v;
    }
}

// ---------------------------------------------------------------------------
// Host launch. Inputs (setup_inputs order, all f32):
//   0:uv 1:feat_2d(shape-only) 2:feat_3d 3:w1 4:b1 5:w2 6:b2 7:w_out 8:b_out
// Workspace: [knn_idx: BS*HW*3 ints][final: BS*C3D*HW floats]
// ---------------------------------------------------------------------------
extern "C" void kernel_launch(void* const* d_in, const int* in_sizes, int n_in,
                              void* d_out, int out_size, void* d_ws, size_t ws_size,
                              hipStream_t stream) {
    const float* uv     = (const float*)d_in[0];
    const float* feat3d = (const float*)d_in[2];
    const float* w1     = (const float*)d_in[3];
    const float* b1     = (const float*)d_in[4];
    const float* w2     = (const float*)d_in[5];
    const float* b2     = (const float*)d_in[6];
    const float* w_out  = (const float*)d_in[7];
    const float* b_out  = (const float*)d_in[8];
    float* out = (float*)d_out;

    int*   knn_idx = (int*)d_ws;
    float* fin     = (float*)((char*)d_ws + (size_t)BS * HW * KNN * sizeof(int)); // 288KB, 256B aligned

    // 1) KNN: 24576 queries, 256 threads/block
    knn_kernel<<<(BS * HW) / 256, 256, 0, stream>>>(uv, knn_idx);
    // 2) gather + MLP -> final
    mlp_gather_kernel<<<(BS * HW) / 256, 256, 0, stream>>>(uv, feat3d, w1, b1, w2, b2,
                                                           knn_idx, fin);
    // 3) WMMA projection: BS * (C3D/16) * (HW/16) = 6144 waves, 8 waves/block
    proj_wmma_kernel<<<(BS * (C3D / 16) * (HW / 16)) / 8, 256, 0, stream>>>(fin, w_out,
                                                                            b_out, out);
}